// TBasisLayerBase_40716289966668
// MI455X (gfx1250) — compile-verified
//
#include <hip/hip_runtime.h>
#include <hip/hip_bf16.h>

typedef __attribute__((ext_vector_type(2))) float v2f;
typedef __attribute__((ext_vector_type(8))) float v8f;

#define NUM_MODES 13
#define RANK 8
#define NUM_BASIS 32

// workspace layout (in floats)
#define CORES_OFF 0                    // 13*256 = 3328 (padded to 4096)
#define A_OFF     4096                 // 4096*64 = 262144  (WMMA-swizzled A fragments)
#define B_OFF     (4096 + 262144)      // 64*16384 = 1048576 (WMMA-swizzled, Morton-permuted B)

// ---------------------------------------------------------------------------
// cores[m][a][d][c] = ra[m][a] * sum_n bw[m][n] * basis[n][a][d][c]
// ---------------------------------------------------------------------------
__global__ void tr_cores_kernel(const float* __restrict__ bw,
                                const float* __restrict__ basis,
                                const float* __restrict__ ra,
                                float* __restrict__ cores) {
  int t = blockIdx.x * blockDim.x + threadIdx.x;      // 13*256 = 3328 exact
  int m   = t >> 8;       // mode
  int rem = t & 255;      // a*32 + d*8 + c
  int a   = rem >> 5;
  float s = 0.f;
  #pragma unroll
  for (int n = 0; n < NUM_BASIS; ++n)
    s += bw[m * NUM_BASIS + n] * basis[n * 256 + rem];
  cores[t] = s * ra[m * RANK + a];
}

// ---------------------------------------------------------------------------
// A[i][k=a*8+b] = L[a,i,b] = row_a( G0[d0] G1[d1] ... G5[d5] )[b]
// stored pre-swizzled into WMMA 16x4-f32 A-fragment layout:
//   float index = ((it*16 + q)*32 + lane)*2 + e,  k = 4q + (lane>=16?2:0) + e
// ---------------------------------------------------------------------------
__global__ void tr_prepA_kernel(const float* __restrict__ cores,
                                float* __restrict__ A) {
  int t = blockIdx.x * blockDim.x + threadIdx.x;      // 8*4096 = 32768 exact
  int a = t >> 12;
  int i = t & 4095;
  float v[8], nv[8];
  int d0 = (i >> 10) & 3;
  #pragma unroll
  for (int c = 0; c < 8; ++c) v[c] = cores[a * 32 + d0 * 8 + c];
  #pragma unroll
  for (int m = 1; m < 6; ++m) {
    int dm = (i >> (2 * (5 - m))) & 3;
    const float* G = cores + m * 256 + dm * 8;        // G[b][c] at G[b*32+c]
    #pragma unroll
    for (int c = 0; c < 8; ++c) {
      float s = 0.f;
      #pragma unroll
      for (int b = 0; b < 8; ++b) s += v[b] * G[b * 32 + c];
      nv[c] = s;
    }
    #pragma unroll
    for (int c = 0; c < 8; ++c) v[c] = nv[c];
  }
  int it = i >> 4, mr = i & 15;
  #pragma unroll
  for (int b = 0; b < 8; ++b) {
    int k = a * 8 + b;
    int q = k >> 2, e = k & 1;
    int lane = mr + (((k & 3) >= 2) ? 16 : 0);
    A[(size_t)((it * 16 + q) * 32 + lane) * 2 + e] = v[b];
  }
}

// ---------------------------------------------------------------------------
// B[k=a*8+b][p] = R[b, morton(p), a],  p = rin*128 + cin, j = interleave(rin,cin)
// stored pre-swizzled into WMMA 4x16-f32 B-fragment layout:
//   float index = ((pt*16 + q)*32 + lane)*2 + e,  k = 4q + (lane>=16?2:0) + e
// ---------------------------------------------------------------------------
__global__ void tr_prepB_kernel(const float* __restrict__ cores,
                                float* __restrict__ B) {
  int t = blockIdx.x * blockDim.x + threadIdx.x;      // 8*16384 = 131072 exact
  int b = t >> 14;
  int p = t & 16383;
  int rin = p >> 7, cin = p & 127;
  int j = 0;
  #pragma unroll
  for (int s = 0; s < 7; ++s) {
    j |= ((rin >> (6 - s)) & 1) << (13 - 2 * s);
    j |= ((cin >> (6 - s)) & 1) << (12 - 2 * s);
  }
  float v[8], nv[8];
  int d6 = (j >> 12) & 3;
  #pragma unroll
  for (int c = 0; c < 8; ++c) v[c] = cores[6 * 256 + b * 32 + d6 * 8 + c];
  #pragma unroll
  for (int m = 7; m < 13; ++m) {
    int dm = (j >> (2 * (12 - m))) & 3;
    const float* G = cores + m * 256 + dm * 8;
    #pragma unroll
    for (int c = 0; c < 8; ++c) {
      float s = 0.f;
      #pragma unroll
      for (int bb = 0; bb < 8; ++bb) s += v[bb] * G[bb * 32 + c];
      nv[c] = s;
    }
    #pragma unroll
    for (int c = 0; c < 8; ++c) v[c] = nv[c];
  }
  int pt = p >> 4, pl = p & 15;
  #pragma unroll
  for (int a = 0; a < 8; ++a) {
    int k = a * 8 + b;
    int q = k >> 2, e = k & 1;
    int lane = pl + (((k & 3) >= 2) ? 16 : 0);
    B[(size_t)((pt * 16 + q) * 32 + lane) * 2 + e] = v[a];
  }
}

// ---------------------------------------------------------------------------
// C = A(4096x64) @ B(64x16384) via v_wmma_f32_16x16x4_f32; each wave does a
// 16x128 slab = 16 contiguous 512B row segments of sixteen 128x128 z-order
// output blocks. Block/row offsets constant-fold into immediate store offsets.
// ---------------------------------------------------------------------------
__global__ void __launch_bounds__(256)
tr_gemm_kernel(const float* __restrict__ A,
               const float* __restrict__ B,
               float* __restrict__ out) {
  int tid   = blockIdx.x * blockDim.x + threadIdx.x;
  int wave  = tid >> 5;            // 0..32767
  int lane  = tid & 31;
  int it    = wave >> 7;           // i-tile 0..255 (rows i = it*16 .. +15)
  int pt8   = wave & 127;          // row-in-block rin (cols p = pt8*128 .. +127)
  int laneHi = lane >> 4;
  int lanelo = lane & 15;

  // A fragments: 16 quads, resident for all 8 n-tiles
  const v2f* Af = (const v2f*)(A + (size_t)it * 1024);
  v2f af[16];
  #pragma unroll
  for (int q = 0; q < 16; ++q) af[q] = Af[q * 32 + lane];

  v8f acc[8];
  #pragma unroll
  for (int t = 0; t < 8; ++t) {
    const v2f* Bf = (const v2f*)(B + (size_t)(pt8 * 8 + t) * 1024);
    v8f c = {0.f, 0.f, 0.f, 0.f, 0.f, 0.f, 0.f, 0.f};
    #pragma unroll
    for (int q = 0; q < 16; ++q) {
      v2f bf = Bf[q * 32 + lane];
      c = __builtin_amdgcn_wmma_f32_16x16x4_f32(
          /*neg_a=*/false, af[q], /*neg_b=*/false, bf,
          /*c_mod=*/(short)0, c, /*reuse_a=*/false, /*reuse_b=*/false);
    }
    acc[t] = c;
  }

  // it bits (r0 c0 r1 c1 r2 c2 r3 c3) -> high block coords
  unsigned u = (unsigned)it;
  unsigned BRhi = ((u >> 7) & 1) * 8 + ((u >> 5) & 1) * 4 + ((u >> 3) & 1) * 2 + ((u >> 1) & 1);
  unsigned BChi = ((u >> 6) & 1) * 8 + ((u >> 4) & 1) * 4 + ((u >> 2) & 1) * 2 + (u & 1);
  // row = (BRhi*4 + laneHi*2 + r5)*128 + pt8 ; col = (BChi*4 + c4*2 + c5)*128 + t*16 + lanelo
  size_t common = ((size_t)(BRhi * 4u + (unsigned)laneHi * 2u) * 128u + (unsigned)pt8) * 8192u
                + (size_t)BChi * 512u + (unsigned)lanelo;

  #pragma unroll
  for (int t = 0; t < 8; ++t) {
    #pragma unroll
    for (int e = 0; e < 8; ++e) {
      // m = e + 8*laneHi : r5 = (e>>1)&1, c4 = (e>>2)&1, c5 = e&1
      size_t cte = ((size_t)((e >> 1) & 1) << 20)   // r5 -> +128 rows
                 + (size_t)(((e >> 2) & 1) * 256)   // c4 -> +256 cols
                 + (size_t)((e & 1) * 128)          // c5 -> +128 cols
                 + (size_t)(t * 16);
      out[common + cte] = acc[t][e];
    }
  }
}

// ---------------------------------------------------------------------------
extern "C" void kernel_launch(void* const* d_in, const int* in_sizes, int n_in,
                              void* d_out, int out_size, void* d_ws, size_t ws_size,
                              hipStream_t stream) {
  const float* bw    = (const float*)d_in[0];   // (13,32)
  const float* basis = (const float*)d_in[1];   // (32,8,4,8)
  const float* ra    = (const float*)d_in[2];   // (13,8)
  float* out = (float*)d_out;                   // (8192,8192) f32
  float* ws  = (float*)d_ws;

  float* cores = ws + CORES_OFF;
  float* Abuf  = ws + A_OFF;
  float* Bbuf  = ws + B_OFF;

  tr_cores_kernel<<<13,   256, 0, stream>>>(bw, basis, ra, cores);
  tr_prepA_kernel<<<128,  256, 0, stream>>>(cores, Abuf);
  tr_prepB_kernel<<<512,  256, 0, stream>>>(cores, Bbuf);
  tr_gemm_kernel <<<4096, 256, 0, stream>>>(Abuf, Bbuf, out);
}